// SAGPooling_5222680232121
// MI455X (gfx1250) — compile-verified
//
#include <hip/hip_runtime.h>
#include <hip/hip_bf16.h>
#include <stdint.h>

#define B_GRAPHS 64
#define C_IN     256
#define NPG      4096
#define DEG      16
#define N_NODES  (B_GRAPHS * NPG)   // 262144
#define E_EDGES  (N_NODES * DEG)    // 4194304
#define K_TOP    (NPG / 2)          // 2048
#define BK       (B_GRAPHS * K_TOP) // 131072

typedef float v2f __attribute__((ext_vector_type(2)));
typedef float v8f __attribute__((ext_vector_type(8)));

// ---------------------------------------------------------------------------
// Kernel 1: fused projections s_rel = x@W_rel, s_root = x@W_root via
// V_WMMA_F32_16X16X4_F32. One wave computes a 16-row tile of x.
// B operand comes from a zero-padded [N=16][K=256] LDS image of the weights:
// row 0 = W_rel, row 1 = W_root, rows 2..15 = 0.  This makes the per-chunk
// operand build branch-free: one global b64 load (A) + one ds b64 load (B)
// per WMMA, no exec-mask games.
// Block = 256 threads = 8 waves = 128 rows.
// ---------------------------------------------------------------------------
__global__ void proj_wmma_kernel(const float* __restrict__ x,
                                 const float* __restrict__ W_rel,
                                 const float* __restrict__ W_root,
                                 float* __restrict__ s_rel,
                                 float* __restrict__ s_root) {
    __shared__ __align__(16) float Bpad[16 * C_IN];   // [column n][k]
    const int tid = threadIdx.x;
    for (int i = tid; i < 16 * C_IN; i += 256) {
        int n = i >> 8;           // output column (0=rel, 1=root, 2..15 pad)
        int k = i & (C_IN - 1);
        float v = 0.0f;
        if (n == 0)      v = W_rel[k];
        else if (n == 1) v = W_root[k];
        Bpad[i] = v;
    }
    __syncthreads();

    const int lane    = tid & 31;
    const int wave    = tid >> 5;
    const int rowBase = blockIdx.x * 128 + wave * 16;
    const int l16     = lane & 15;
    const int half    = lane >> 4;     // 0 -> K pair {0,1}, 1 -> K pair {2,3}
    const float* xr   = x + (size_t)(rowBase + l16) * C_IN;
    const float* bp   = &Bpad[l16 * C_IN];

    v8f c = {};
#pragma unroll 8
    for (int kb = 0; kb < C_IN; kb += 4) {
        const int kk = kb + half * 2;
        // A (16x4): lane L<16 -> row L, K = kb..kb+1 ; lane L>=16 -> K = kb+2..kb+3
        v2f a = *(const v2f*)(xr + kk);
        // B (4x16): lane L<16 -> (K=kb..kb+1, N=L) ; lane L>=16 -> (K=kb+2..3, N=L-16)
        v2f b = *(const v2f*)(bp + kk);
        c = __builtin_amdgcn_wmma_f32_16x16x4_f32(
                /*neg_a=*/false, a, /*neg_b=*/false, b,
                /*c_mod=*/(short)0, c, /*reuse_a=*/false, /*reuse_b=*/false);
    }

    // D layout: VGPR r, lanes 0-15 -> (M=r, N=lane); lanes 16-31 -> (M=8+r, N=lane-16)
    if (l16 == 0) {               // column 0 = s_rel
        float* outp = s_rel + rowBase + half * 8;
#pragma unroll
        for (int r = 0; r < 8; ++r) outp[r] = c[r];
    } else if (l16 == 1) {        // column 1 = s_root
        float* outp = s_root + rowBase + half * 8;
#pragma unroll
        for (int r = 0; r < 8; ++r) outp[r] = c[r];
    }
}

// ---------------------------------------------------------------------------
// Init: agg = 0, node_map = -1
// ---------------------------------------------------------------------------
__global__ void init_kernel(float* __restrict__ agg, int* __restrict__ node_map) {
    int i = blockIdx.x * blockDim.x + threadIdx.x;
    if (i < N_NODES) { agg[i] = 0.0f; node_map[i] = -1; }
}

// ---------------------------------------------------------------------------
// Segment sum: agg[dst[e]] += s_rel[src[e]]
// ---------------------------------------------------------------------------
__global__ void segsum_kernel(const int* __restrict__ src, const int* __restrict__ dst,
                              const float* __restrict__ s_rel, float* __restrict__ agg) {
    int e = blockIdx.x * blockDim.x + threadIdx.x;
    if (e < E_EDGES) atomicAdd(&agg[dst[e]], s_rel[src[e]]);
}

// ---------------------------------------------------------------------------
// score = tanh(agg + b_rel + s_root)
// ---------------------------------------------------------------------------
__global__ void score_kernel(const float* __restrict__ agg, const float* __restrict__ s_root,
                             const float* __restrict__ b_rel, float* __restrict__ score) {
    int i = blockIdx.x * blockDim.x + threadIdx.x;
    if (i < N_NODES) score[i] = tanhf(agg[i] + b_rel[0] + s_root[i]);
}

// ---------------------------------------------------------------------------
// Per-graph top-k via full bitonic sort of 4096 64-bit keys in LDS.
// key = (flipped-float-bits(score) , index): ascending u64 sort gives
// descending score with lower-index-first tie break (matches lax.top_k).
// One block (256 threads) per graph.
// ---------------------------------------------------------------------------
__global__ void topk_kernel(const float* __restrict__ score,
                            int* __restrict__ perm, float* __restrict__ score_perm) {
    __shared__ unsigned long long keys[NPG];
    const int b = blockIdx.x;
    const int tid = threadIdx.x;
    const float* sg = score + (size_t)b * NPG;

    for (int i = tid; i < NPG; i += blockDim.x) {
        unsigned int bits = __float_as_uint(sg[i]);
        unsigned int asc  = (bits & 0x80000000u) ? ~bits : (bits | 0x80000000u);
        unsigned int desc = ~asc;   // ascending in desc == descending in score
        keys[i] = ((unsigned long long)desc << 32) | (unsigned int)i;
    }
    __syncthreads();

    for (int size = 2; size <= NPG; size <<= 1) {
        for (int stride = size >> 1; stride > 0; stride >>= 1) {
            for (int i = tid; i < NPG; i += blockDim.x) {
                int j = i ^ stride;
                if (j > i) {
                    bool dir_up = ((i & size) == 0);
                    unsigned long long a = keys[i], bb = keys[j];
                    bool swap = dir_up ? (a > bb) : (a < bb);
                    if (swap) { keys[i] = bb; keys[j] = a; }
                }
            }
            __syncthreads();
        }
    }

    for (int j = tid; j < K_TOP; j += blockDim.x) {
        int idx  = (int)(keys[j] & 0xFFFFFFFFu);
        int node = b * NPG + idx;
        perm[b * K_TOP + j]       = node;
        score_perm[b * K_TOP + j] = sg[idx];
    }
}

// ---------------------------------------------------------------------------
// node_map[perm[j]] = j  +  small float outputs (batch_out, perm, score_perm)
// ---------------------------------------------------------------------------
__global__ void scatter_small_kernel(const int* __restrict__ perm,
                                     const float* __restrict__ score_perm,
                                     int* __restrict__ node_map,
                                     float* __restrict__ out_batch,
                                     float* __restrict__ out_perm,
                                     float* __restrict__ out_scoreperm) {
    int j = blockIdx.x * blockDim.x + threadIdx.x;
    if (j < BK) {
        int p = perm[j];
        node_map[p]       = j;
        out_batch[j]      = (float)(p >> 12);   // p / NPG
        out_perm[j]       = (float)p;
        out_scoreperm[j]  = score_perm[j];
    }
}

// ---------------------------------------------------------------------------
// x_out[j,:] = x[perm[j],:] * score_perm[j]   (float4 vectorized)
// ---------------------------------------------------------------------------
__global__ void gather_kernel(const float* __restrict__ x, const int* __restrict__ perm,
                              const float* __restrict__ score_perm,
                              float* __restrict__ x_out) {
    int t = blockIdx.x * blockDim.x + threadIdx.x;     // BK * (C/4) threads
    if (t >= BK * (C_IN / 4)) return;
    int row  = t >> 6;              // C/4 = 64 quads per row
    int col4 = (t & 63) << 2;
    int p    = perm[row];
    float s  = score_perm[row];
    float4 v = *(const float4*)(x + (size_t)p * C_IN + col4);
    float4 o = make_float4(v.x * s, v.y * s, v.z * s, v.w * s);
    *(float4*)(x_out + (size_t)row * C_IN + col4) = o;
}

// ---------------------------------------------------------------------------
// Edge remap: nr/nc via node_map; mask; -1 for dropped edges (as floats)
// ---------------------------------------------------------------------------
__global__ void edge_kernel(const int* __restrict__ src, const int* __restrict__ dst,
                            const int* __restrict__ node_map,
                            float* __restrict__ out_ei, float* __restrict__ out_mask) {
    int e = blockIdx.x * blockDim.x + threadIdx.x;
    if (e < E_EDGES) {
        int nr = node_map[src[e]];
        int nc = node_map[dst[e]];
        bool m = (nr >= 0) && (nc >= 0);
        out_ei[e]           = m ? (float)nr : -1.0f;
        out_ei[E_EDGES + e] = m ? (float)nc : -1.0f;
        out_mask[e]         = m ? 1.0f : 0.0f;
    }
}

// ---------------------------------------------------------------------------
extern "C" void kernel_launch(void* const* d_in, const int* in_sizes, int n_in,
                              void* d_out, int out_size, void* d_ws, size_t ws_size,
                              hipStream_t stream) {
    const float* x          = (const float*)d_in[0];
    const int*   edge_index = (const int*)  d_in[1];
    // d_in[2] = batch (derived instead: perm / NPG)
    const float* W_rel      = (const float*)d_in[3];
    const float* b_rel      = (const float*)d_in[4];
    const float* W_root     = (const float*)d_in[5];
    const int* src = edge_index;             // row 0 of [2,E]
    const int* dst = edge_index + E_EDGES;   // row 1

    // workspace layout
    float* s_rel      = (float*)d_ws;               // N
    float* s_root     = s_rel  + N_NODES;           // N
    float* agg        = s_root + N_NODES;           // N
    float* score      = agg    + N_NODES;           // N
    float* score_perm = score  + N_NODES;           // BK
    int*   perm       = (int*)(score_perm + BK);    // BK
    int*   node_map   = perm + BK;                  // N

    // output layout (all float, reference return order)
    float* out            = (float*)d_out;
    float* out_x          = out;                               // BK*C
    float* out_ei         = out_x   + (size_t)BK * C_IN;       // 2*E
    float* out_mask       = out_ei  + 2 * (size_t)E_EDGES;     // E
    float* out_batch      = out_mask + (size_t)E_EDGES;        // BK
    float* out_perm       = out_batch + BK;                    // BK
    float* out_scoreperm  = out_perm  + BK;                    // BK

    init_kernel         <<<(N_NODES + 255) / 256, 256, 0, stream>>>(agg, node_map);
    proj_wmma_kernel    <<<N_NODES / 128,        256, 0, stream>>>(x, W_rel, W_root, s_rel, s_root);
    segsum_kernel       <<<(E_EDGES + 255) / 256, 256, 0, stream>>>(src, dst, s_rel, agg);
    score_kernel        <<<(N_NODES + 255) / 256, 256, 0, stream>>>(agg, s_root, b_rel, score);
    topk_kernel         <<<B_GRAPHS,             256, 0, stream>>>(score, perm, score_perm);
    scatter_small_kernel<<<(BK + 255) / 256,     256, 0, stream>>>(perm, score_perm, node_map,
                                                                   out_batch, out_perm, out_scoreperm);
    gather_kernel       <<<(BK * (C_IN / 4) + 255) / 256, 256, 0, stream>>>(x, perm, score_perm, out_x);
    edge_kernel         <<<(E_EDGES + 255) / 256, 256, 0, stream>>>(src, dst, node_map, out_ei, out_mask);
}